// AdjacencyProjector_11957188952590
// MI455X (gfx1250) — compile-verified
//
#include <hip/hip_runtime.h>

typedef float v2f __attribute__((ext_vector_type(2)));
typedef float v4f __attribute__((ext_vector_type(4)));
typedef float v8f __attribute__((ext_vector_type(8)));

#define B_DIM 4
#define N_DIM 4096
#define D_DIM 128

// ---------------------------------------------------------------------------
// Phase 1: a[r] = E[r,:]·w_i, b[r] = E[r,:]·w_j for r in [0, B*N)
// via chained V_WMMA_F32_16X16X4_F32: C(16x16) += A(16x4) x Bw(4x16),
// 32 K-steps covering K=128. Column 0 of C accumulates a, column 1 b.
// One wave handles 16 rows; block = 256 threads = 8 waves = 128 rows.
// Inner loop is fully branchless so the 32 steps pipeline (no exec-mask
// predicated loads, no per-iteration s_wait_loadcnt 0 stalls).
// ---------------------------------------------------------------------------
__global__ __launch_bounds__(256) void rowdots_wmma(const float* __restrict__ E,
                                                    const float* __restrict__ W,
                                                    float* __restrict__ aOut,
                                                    float* __restrict__ bOut) {
    const int lane    = threadIdx.x & 31;
    const int wave    = threadIdx.x >> 5;
    const int baseRow = blockIdx.x * 128 + wave * 16;

    const int colSel = lane & 15;         // A: row M = colSel ; B/C: column N = colSel
    const int kHalf  = (lane >> 4) << 1;  // lanes 0-15 -> K+{0,1}, lanes 16-31 -> K+{2,3}

    const float* __restrict__ rowPtr = E + (size_t)(baseRow + colSel) * D_DIM;

    // Per-lane W pointer: column 0 -> w_i, column 1 -> w_j, others read w_i but
    // are zeroed by bmask (finite values, so 0*x == 0). Unconditional loads =>
    // no exec-mask predication in the loop.
    const float* __restrict__ wsel = W + ((colSel == 1) ? D_DIM : 0);
    const float bmask = (colSel < 2) ? 1.0f : 0.0f;

    v8f c = {};  // C/D accumulator: VGPRg = rows {g, g+8}, cols striped over lanes

    #pragma unroll
    for (int k = 0; k < 32; ++k) {
        const int koff = 4 * k + kHalf;  // even -> 8B-aligned v2f loads

        // A 16x4 f32: lane L(<16): M=L, v0=K0,v1=K1 ; lane 16+L: M=L, v0=K2,v1=K3
        v2f a = *(const v2f*)(rowPtr + koff);

        // B 4x16 f32: v0 = row K=koff, v1 = row K=koff+1, columns over lanes.
        v2f bw = *(const v2f*)(wsel + koff);
        v2f b;
        b.x = bw.x * bmask;
        b.y = bw.y * bmask;

        // (neg_a, A, neg_b, B, c_mod, C, reuse_a, reuse_b)
        c = __builtin_amdgcn_wmma_f32_16x16x4_f32(false, a, false, b,
                                                  (short)0, c, false, false);
    }

    // C[m][0] = a-row m, C[m][1] = b-row m.
    // VGPRg: lanes 0-15 hold row M=g, lanes 16-31 hold row M=g+8, column = lane&15.
    if (lane == 0) {
        #pragma unroll
        for (int g = 0; g < 8; ++g) aOut[baseRow + g] = c[g];
    } else if (lane == 16) {
        #pragma unroll
        for (int g = 0; g < 8; ++g) aOut[baseRow + 8 + g] = c[g];
    } else if (lane == 1) {
        #pragma unroll
        for (int g = 0; g < 8; ++g) bOut[baseRow + g] = c[g];
    } else if (lane == 17) {
        #pragma unroll
        for (int g = 0; g < 8; ++g) bOut[baseRow + 8 + g] = c[g];
    }
}

// ---------------------------------------------------------------------------
// Phase 2: out[b][i][j] = a[b*N+i] + bv[b*N+j]  (268 MB streaming store).
// Each block covers one quarter of one output row (256 threads x float4 = 1024
// floats). row is blockIdx-derived -> a[row] is a scalar (SGPR) load. Output
// written with non-temporal B128 stores (never re-read, > L2 capacity).
// ---------------------------------------------------------------------------
__global__ __launch_bounds__(256) void outer_add(const float* __restrict__ a,
                                                 const float* __restrict__ bv,
                                                 v4f* __restrict__ out) {
    const size_t row = (size_t)(blockIdx.x >> 2);               // b*N + i, uniform per block
    const int    j4  = ((blockIdx.x & 3) << 8) | threadIdx.x;   // float4 index within row

    const float av = a[row];
    const int   b  = (int)(row >> 12);                          // row / N

    const v4f* __restrict__ bvec = (const v4f*)bv + ((size_t)b << 10);  // b*N/4
    v4f x = bvec[j4];
    v4f r;
    r.x = av + x.x;
    r.y = av + x.y;
    r.z = av + x.z;
    r.w = av + x.w;

    __builtin_nontemporal_store(r, out + (row << 10) + j4);
}

extern "C" void kernel_launch(void* const* d_in, const int* in_sizes, int n_in,
                              void* d_out, int out_size, void* d_ws, size_t ws_size,
                              hipStream_t stream) {
    const float* E = (const float*)d_in[0];   // (4, 4096, 128) f32
    const float* W = (const float*)d_in[1];   // (1, 256) f32
    float* out  = (float*)d_out;              // (4, 4096, 4096) f32
    float* aBuf = (float*)d_ws;               // B*N floats
    float* bBuf = aBuf + (size_t)B_DIM * N_DIM;

    // Phase 1: 16384 rows, 128 rows per block -> 128 blocks
    rowdots_wmma<<<128, 256, 0, stream>>>(E, W, aBuf, bBuf);

    // Phase 2: B*N rows, 4 blocks per row -> 65536 blocks
    const int blocks = B_DIM * N_DIM * 4;
    outer_add<<<blocks, 256, 0, stream>>>(aBuf, bBuf, (v4f*)out);
}